// seq2seq_37039797960768
// MI455X (gfx1250) — compile-verified
//
#include <hip/hip_runtime.h>
#include <stdint.h>

// Problem constants (match reference)
#define V_ 32000
#define E_ 256
#define U_ 512
#define B_ 32
#define S_ 64
#define T_ 25

typedef __attribute__((ext_vector_type(2))) float v2f;
typedef __attribute__((ext_vector_type(8))) float v8f;

// CDNA5 async global->LDS path (guarded: falls back to vmem+ds if absent)
#if defined(__has_builtin)
#if __has_builtin(__builtin_amdgcn_global_load_async_to_lds_b32) && \
    __has_builtin(__builtin_amdgcn_s_wait_asynccnt)
#define USE_ASYNC_LDS 1
#endif
#endif
#ifndef USE_ASYNC_LDS
#define USE_ASYNC_LDS 0
#endif

// Builtin operand types (from hipcc diagnostic: params are int* in AS(1)/AS(3))
typedef __attribute__((address_space(1))) int* as1ip;
typedef __attribute__((address_space(3))) int* as3ip;

__device__ __forceinline__ float sigmoidf_(float x) {
  return 1.0f / (1.0f + __expf(-x));
}

// out[i*ncols + c] = emb[idx[i]*ncols + c]
__global__ void k_gather(const float* __restrict__ emb, const int* __restrict__ idx,
                         float* __restrict__ out, int nrows, int ncols) {
  int i = blockIdx.x * blockDim.x + threadIdx.x;
  int total = nrows * ncols;
  if (i >= total) return;
  int r = i / ncols, c = i - r * ncols;
  out[i] = emb[idx[r] * ncols + c];
}

__global__ void k_zero(float* __restrict__ p, int n) {
  int i = blockIdx.x * blockDim.x + threadIdx.x;
  if (i < n) p[i] = 0.0f;
}

// Generic WMMA GEMM, compile-time strides so k-offsets fold into load IOFFSET.
// C[M,N] = A[M,K] @ B[K,N] + bias[N]; one wave per 16x16 C tile.
template <int K, int LDA, int LDB>
__global__ void k_gemm_wmma(const float* __restrict__ A,
                            const float* __restrict__ Bm,
                            float* __restrict__ C, int ldc,
                            const float* __restrict__ bias,
                            int MT, int NT) {
  int wid  = (blockIdx.x * blockDim.x + threadIdx.x) >> 5;
  int lane = threadIdx.x & 31;
  if (wid >= MT * NT) return;            // wave-uniform exit; EXEC stays all-1 for WMMA
  int tm = wid / NT, tn = wid - tm * NT;
  int half = lane >> 4;
  int l    = lane & 15;
  const float* ap = A + (size_t)(tm * 16 + l) * LDA + 2 * half;
  const float* bp = Bm + (size_t)(2 * half) * LDB + tn * 16 + l;
  v8f acc = {};
#pragma unroll 8
  for (int k = 0; k < K; k += 4) {
    v2f a, b;
    a.x = ap[k];
    a.y = ap[k + 1];
    b.x = bp[(size_t)k * LDB];
    b.y = bp[(size_t)k * LDB + LDB];
    acc = __builtin_amdgcn_wmma_f32_16x16x4_f32(false, a, false, b, (short)0, acc,
                                                false, false);
  }
  int col = tn * 16 + l;
  float bv = bias[col];
  int mbase = tm * 16 + half * 8;        // C: VGPR i -> M = i + 8*half
#pragma unroll
  for (int i = 0; i < 8; ++i)
    C[(size_t)(mbase + i) * ldc + col] = acc[i] + bv;
}

// Dominant GEMM: logits = h[32,512] @ fc_W[512,32000] + fc_b, C strided into out[B,T,V].
// Workgroup = 8 waves = 2x4 WMMA tiles (32 rows x 64 cols per slab), grid-stride slabs.
// A is staged TRANSPOSED into LDS (At[k][m], 64KB, bank-conflict-free) once per block,
// via async global->LDS copies when available.
__global__ void k_fc_gemm(const float* __restrict__ A,   // [32,512]
                          const float* __restrict__ Bm,  // [512, 32000]
                          float* __restrict__ C,         // out + t*V, row stride T*V
                          const float* __restrict__ bias) {
  __shared__ float At[U_ * B_];          // [512][32] transposed, exactly 64KB
#if USE_ASYNC_LDS
  for (int i = threadIdx.x; i < B_ * U_; i += 256) {
    int m = i >> 9;                      // i / 512
    int k = i & (U_ - 1);
    const float* gp = A + i;
    float* lp = &At[k * B_ + m];
    __builtin_amdgcn_global_load_async_to_lds_b32(
        (as1ip)(uintptr_t)gp,
        (as3ip)(uint32_t)(uintptr_t)lp,
        0, 0);
  }
  __builtin_amdgcn_s_wait_asynccnt(0);
#else
  for (int i = threadIdx.x; i < B_ * U_; i += 256) {
    int m = i >> 9;
    int k = i & (U_ - 1);
    At[k * B_ + m] = A[i];
  }
#endif
  __syncthreads();

  int wave = threadIdx.x >> 5;
  int lane = threadIdx.x & 31;
  int half = lane >> 4, l = lane & 15;
  int tm  = wave >> 2;                   // 0..1  (M tile)
  int tn4 = wave & 3;                    // 0..3  (N tile within slab)
  const float* atp = At + tm * 16 + l + (size_t)(2 * half) * B_;
  const int nslab = V_ / 64;             // 500
  for (int slab = blockIdx.x; slab < nslab; slab += gridDim.x) {
    int col = slab * 64 + tn4 * 16 + l;
    const float* bp = Bm + (size_t)(2 * half) * V_ + col;
    v8f acc = {};
#pragma unroll 8
    for (int k = 0; k < U_; k += 4) {
      v2f a, b;
      a.x = atp[k * B_];                 // ds_load, immediate offsets
      a.y = atp[k * B_ + B_];
      b.x = bp[(size_t)k * V_];
      b.y = bp[(size_t)k * V_ + V_];
      acc = __builtin_amdgcn_wmma_f32_16x16x4_f32(false, a, false, b, (short)0, acc,
                                                  false, false);
    }
    float bv = bias[col];
    int mbase = tm * 16 + half * 8;
#pragma unroll
    for (int i = 0; i < 8; ++i)
      C[(size_t)(mbase + i) * (T_ * V_) + col] = acc[i] + bv;
  }
}

// One encoder GRU step: gh = h@Wh + bh (3 fused gate tiles), gate with precomputed
// gx (= x@Wi + bi), write h_out and enc_out[:,t,:]. Wave computes a 16x16 slice of U.
__global__ void k_enc_step(const float* __restrict__ h_in,
                           const float* __restrict__ Wh,   // [U, 3U]
                           const float* __restrict__ bh,   // [3U]
                           const float* __restrict__ gx_t, // gx + t*3U, row stride S*3U
                           float* __restrict__ h_out,      // [B,U]
                           float* __restrict__ out_t) {    // enc_out + t*U, row stride S*U
  const int gx_ld = S_ * 3 * U_;
  const int out_ld = S_ * U_;
  int wid  = (blockIdx.x * blockDim.x + threadIdx.x) >> 5;
  int lane = threadIdx.x & 31;
  const int nt = U_ / 16, mt = B_ / 16;
  if (wid >= mt * nt) return;
  int tm = wid / nt, tn = wid - tm * nt;
  int half = lane >> 4, l = lane & 15;
  int n   = tn * 16 + l;
  v8f az = {}, ar = {}, ac = {};
  const float* ap = h_in + (size_t)(tm * 16 + l) * U_ + 2 * half;
  const float* wp = Wh + (size_t)(2 * half) * (3 * U_) + n;
#pragma unroll 4
  for (int k = 0; k < U_; k += 4) {
    v2f a, bz, br, bc;
    a.x = ap[k];
    a.y = ap[k + 1];
    const int r0 = k * (3 * U_);
    const int r1 = r0 + 3 * U_;
    bz.x = wp[r0];            bz.y = wp[r1];
    br.x = wp[r0 + U_];       br.y = wp[r1 + U_];
    bc.x = wp[r0 + 2 * U_];   bc.y = wp[r1 + 2 * U_];
    az = __builtin_amdgcn_wmma_f32_16x16x4_f32(false, a, false, bz, (short)0, az, false, false);
    ar = __builtin_amdgcn_wmma_f32_16x16x4_f32(false, a, false, br, (short)0, ar, false, false);
    ac = __builtin_amdgcn_wmma_f32_16x16x4_f32(false, a, false, bc, (short)0, ac, false, false);
  }
  float bhz = bh[n], bhr = bh[U_ + n], bhh = bh[2 * U_ + n];
  int mbase = tm * 16 + half * 8;
#pragma unroll
  for (int i = 0; i < 8; ++i) {
    int m = mbase + i;
    const float* gxr_ = gx_t + (size_t)m * gx_ld;
    float xz = gxr_[n], xr = gxr_[U_ + n], xh = gxr_[2 * U_ + n];
    float z = sigmoidf_(xz + az[i] + bhz);
    float r = sigmoidf_(xr + ar[i] + bhr);
    float c = tanhf(xh + r * (ac[i] + bhh));
    float hp = h_in[(size_t)m * U_ + n];
    float hn = z * hp + (1.0f - z) * c;
    h_out[(size_t)m * U_ + n]     = hn;
    out_t[(size_t)m * out_ld + n] = hn;
  }
}

// score[b,s] = Vw . tanh(keys[b,s,:] + q[b,:]) + bV   (one wave per (b,s))
__global__ void k_attn_score(const float* __restrict__ keys,  // [B,S,U]
                             const float* __restrict__ q,     // [B,U]
                             const float* __restrict__ Vw,    // [U]
                             const float* __restrict__ bV,    // [1]
                             float* __restrict__ score) {     // [B,S]
  int wid  = (blockIdx.x * blockDim.x + threadIdx.x) >> 5;
  int lane = threadIdx.x & 31;
  if (wid >= B_ * S_) return;
  int b = wid / S_;
  const float* kp = keys + (size_t)wid * U_;
  const float* qp = q + (size_t)b * U_;
  float sum = 0.0f;
  for (int u = lane; u < U_; u += 32)
    sum += tanhf(kp[u] + qp[u]) * Vw[u];
#pragma unroll
  for (int off = 16; off > 0; off >>= 1)
    sum += __shfl_xor(sum, off, 32);
  if (lane == 0) score[wid] = sum + bV[0];
}

// softmax over S per batch (block = S threads)
__global__ void k_softmax(const float* __restrict__ score, float* __restrict__ aw) {
  __shared__ float red[S_];
  int b = blockIdx.x, s = threadIdx.x;
  float v = score[b * S_ + s];
  red[s] = v; __syncthreads();
  for (int off = S_ / 2; off > 0; off >>= 1) {
    if (s < off) red[s] = fmaxf(red[s], red[s + off]);
    __syncthreads();
  }
  float mx = red[0]; __syncthreads();
  float e = __expf(v - mx);
  red[s] = e; __syncthreads();
  for (int off = S_ / 2; off > 0; off >>= 1) {
    if (s < off) red[s] += red[s + off];
    __syncthreads();
  }
  aw[b * S_ + s] = e / red[0];
}

// ctx[b,u] = sum_s aw[b,s] * enc_out[b,s,u]
__global__ void k_ctx(const float* __restrict__ aw, const float* __restrict__ enc_out,
                      float* __restrict__ ctx) {
  int i = blockIdx.x * blockDim.x + threadIdx.x;
  if (i >= B_ * U_) return;
  int b = i / U_, u = i - b * U_;
  const float* eo = enc_out + (size_t)b * (S_ * U_) + u;
  const float* a  = aw + b * S_;
  float s = 0.0f;
#pragma unroll 8
  for (int t = 0; t < S_; ++t) s += a[t] * eo[(size_t)t * U_];
  ctx[i] = s;
}

// g[b, :] = concat(ctx[b,:], dec_ex[b,t,:])
__global__ void k_concat(const float* __restrict__ ctx, const float* __restrict__ exd_t,
                         float* __restrict__ g) {
  const int GW = U_ + E_;
  int i = blockIdx.x * blockDim.x + threadIdx.x;
  if (i >= B_ * GW) return;
  int b = i / GW, c = i - b * GW;
  g[i] = (c < U_) ? ctx[b * U_ + c] : exd_t[(size_t)b * (T_ * E_) + (c - U_)];
}

// Decoder GRU gate from ZERO hidden state (faithful to reference bug):
// gh = 0@Wh + bh = bh  =>  h_new = (1-z)*c  (dec_Wh is dead)
__global__ void k_dec_gate(const float* __restrict__ gx,  // [B,3U] incl. bi
                           const float* __restrict__ bh,  // [3U]
                           float* __restrict__ h) {       // [B,U]
  int i = blockIdx.x * blockDim.x + threadIdx.x;
  if (i >= B_ * U_) return;
  int b = i / U_, u = i - b * U_;
  const float* gr = gx + (size_t)b * 3 * U_;
  float z = sigmoidf_(gr[u] + bh[u]);
  float r = sigmoidf_(gr[U_ + u] + bh[U_ + u]);
  float c = tanhf(gr[2 * U_ + u] + r * bh[2 * U_ + u]);
  h[i] = (1.0f - z) * c;
}

template <int K, int LDA, int LDB>
static inline void launch_gemm_t(const float* A, const float* Bm, float* C, int ldc,
                                 const float* bias, int M, int N, hipStream_t s) {
  int waves = (M / 16) * (N / 16);
  int blocks = (waves * 32 + 255) / 256;   // 8 wave32s per block
  k_gemm_wmma<K, LDA, LDB><<<blocks, 256, 0, s>>>(A, Bm, C, ldc, bias, M / 16, N / 16);
}

extern "C" void kernel_launch(void* const* d_in, const int* in_sizes, int n_in,
                              void* d_out, int out_size, void* d_ws, size_t ws_size,
                              hipStream_t stream) {
  (void)in_sizes; (void)n_in; (void)out_size; (void)ws_size;
  const int*   inp     = (const int*)  d_in[0];
  const int*   tar     = (const int*)  d_in[1];
  const float* enc_emb = (const float*)d_in[2];
  const float* enc_Wi  = (const float*)d_in[3];
  const float* enc_Wh  = (const float*)d_in[4];
  const float* enc_bi  = (const float*)d_in[5];
  const float* enc_bh  = (const float*)d_in[6];
  const float* dec_emb = (const float*)d_in[7];
  const float* dec_Wi  = (const float*)d_in[8];
  /* d_in[9] = dec_Wh is dead: reference runs decoder GRU from zero state */
  const float* dec_bi  = (const float*)d_in[10];
  const float* dec_bh  = (const float*)d_in[11];
  const float* W1      = (const float*)d_in[12];
  const float* b1      = (const float*)d_in[13];
  const float* W2      = (const float*)d_in[14];
  const float* b2      = (const float*)d_in[15];
  const float* Vw      = (const float*)d_in[16];
  const float* bV      = (const float*)d_in[17];
  const float* fc_W    = (const float*)d_in[18];
  const float* fc_b    = (const float*)d_in[19];
  float* out = (float*)d_out;

  // Workspace carve-up (~25 MB)
  char* w = (char*)d_ws;
  size_t off = 0;
  auto alloc = [&](size_t elems) {
    float* p = (float*)(w + off);
    off += ((elems * sizeof(float) + 255) / 256) * 256;
    return p;
  };
  float* ex_enc  = alloc((size_t)B_ * S_ * E_);
  float* gx_enc  = alloc((size_t)B_ * S_ * 3 * U_);
  float* enc_out = alloc((size_t)B_ * S_ * U_);
  float* keys    = alloc((size_t)B_ * S_ * U_);
  float* h0      = alloc((size_t)B_ * U_);
  float* h1      = alloc((size_t)B_ * U_);
  float* ex_dec  = alloc((size_t)B_ * T_ * E_);
  float* qbuf    = alloc((size_t)B_ * U_);
  float* score   = alloc((size_t)B_ * S_);
  float* aw      = alloc((size_t)B_ * S_);
  float* ctx     = alloc((size_t)B_ * U_);
  float* gbuf    = alloc((size_t)B_ * (U_ + E_));
  float* gx_dec  = alloc((size_t)B_ * 3 * U_);

  // 1) Encoder embedding gather + hoisted input GEMM: gx = ex @ Wi + bi  [2048 x 1536]
  {
    int n = B_ * S_ * E_;
    k_gather<<<(n + 255) / 256, 256, 0, stream>>>(enc_emb, inp, ex_enc, B_ * S_, E_);
  }
  launch_gemm_t<E_, E_, 3 * U_>(ex_enc, enc_Wi, gx_enc, 3 * U_, enc_bi,
                                B_ * S_, 3 * U_, stream);
  {
    int n = B_ * U_;
    k_zero<<<(n + 255) / 256, 256, 0, stream>>>(h0, n);
  }

  // 2) Encoder scan: 64 sequential WMMA steps, ping-pong hidden state
  float* hbuf[2] = {h0, h1};
  for (int t = 0; t < S_; ++t) {
    const float* hin = hbuf[t & 1];
    float* hout = hbuf[(t + 1) & 1];
    int waves = (B_ / 16) * (U_ / 16);
    int blocks = (waves * 32 + 255) / 256;
    k_enc_step<<<blocks, 256, 0, stream>>>(hin, enc_Wh, enc_bh,
                                           gx_enc + t * 3 * U_, hout, enc_out + t * U_);
  }
  float* h_dec = hbuf[S_ & 1];  // S_ even -> enc_hidden lands in h0

  // 3) keys = enc_out @ W1 + b1  (hoisted, once)
  launch_gemm_t<U_, U_, U_>(enc_out, W1, keys, U_, b1, B_ * S_, U_, stream);

  // 4) Decoder embedding gather
  {
    int n = B_ * T_ * E_;
    k_gather<<<(n + 255) / 256, 256, 0, stream>>>(dec_emb, tar, ex_dec, B_ * T_, E_);
  }

  // 5) Decoder scan: attention + GRU-from-zero + fc logits (L2-resident fc_W)
  for (int t = 0; t < T_; ++t) {
    launch_gemm_t<U_, U_, U_>(h_dec, W2, qbuf, U_, b2, B_, U_, stream);
    {
      int waves = B_ * S_;
      int blocks = (waves * 32 + 255) / 256;
      k_attn_score<<<blocks, 256, 0, stream>>>(keys, qbuf, Vw, bV, score);
    }
    k_softmax<<<B_, S_, 0, stream>>>(score, aw);
    {
      int n = B_ * U_;
      k_ctx<<<(n + 255) / 256, 256, 0, stream>>>(aw, enc_out, ctx);
    }
    {
      int n = B_ * (U_ + E_);
      k_concat<<<(n + 255) / 256, 256, 0, stream>>>(ctx, ex_dec + t * E_, gbuf);
    }
    launch_gemm_t<U_ + E_, U_ + E_, 3 * U_>(gbuf, dec_Wi, gx_dec, 3 * U_, dec_bi,
                                            B_, 3 * U_, stream);
    {
      int n = B_ * U_;
      k_dec_gate<<<(n + 255) / 256, 256, 0, stream>>>(gx_dec, dec_bh, h_dec);
    }
    // logits[:, t, :] = h_new @ fc_W + fc_b  (grid-stride slabs; A staged in LDS)
    k_fc_gemm<<<125, 256, 0, stream>>>(h_dec, fc_W, out + (size_t)t * V_, fc_b);
  }
}